// ONINorm_9148280340595
// MI455X (gfx1250) — compile-verified
//
#include <hip/hip_runtime.h>
#include <hip/hip_bf16.h>

// ONI norm (Newton-Schulz orthogonalization) for conv weight [2048,1024,3,3]
// g=16 groups, n=128 rows/group, d=9216 cols. All math in f32 via
// V_WMMA_F32_16X16X4_F32 (wave32, A=16x4 -> v2f/lane, B=4x16 -> v2f/lane,
// C/D = v8f/lane). LDS staged in 128x32 chunks, b128 fills, batched
// B-fragment prefetch so WMMAs issue back-to-back.

#define NG 16
#define NN 128
#define DD 9216
#define EPSV 1e-5f
#define TITER 5
#define KSPLIT 16
#define KCHUNK 32
#define APITCH 36                 // 16B-aligned row pitch for A-tiles in LDS
#define MATF (NG * NN * NN)       // floats per batched 128x128 matrix set

typedef __attribute__((ext_vector_type(2))) float v2f;
typedef __attribute__((ext_vector_type(8))) float v8f;

static __device__ __forceinline__ v8f wmma_f32(v2f a, v2f b, v8f c) {
  return __builtin_amdgcn_wmma_f32_16x16x4_f32(
      /*neg_a=*/false, a, /*neg_b=*/false, b,
      /*c_mod=*/(short)0, c, /*reuse_a=*/false, /*reuse_b=*/false);
}

// ---------------------------------------------------------------- row means
__global__ void oni_mean(const float* __restrict__ w, float* __restrict__ mean) {
  __shared__ float red[256];
  const int row = blockIdx.x;
  const float4* p = reinterpret_cast<const float4*>(w + (size_t)row * DD);
  float s = 0.f;
  for (int i = threadIdx.x; i < DD / 4; i += 256) {
    float4 v = p[i];
    s += (v.x + v.y) + (v.z + v.w);
  }
  red[threadIdx.x] = s;
  __syncthreads();
  for (int off = 128; off > 0; off >>= 1) {
    if (threadIdx.x < off) red[threadIdx.x] += red[threadIdx.x + off];
    __syncthreads();
  }
  if (threadIdx.x == 0) mean[row] = red[0] * (1.0f / (float)DD);
}

// ---------------------------------------------------------------- zero S
__global__ void oni_zero(float* __restrict__ S) {
  S[blockIdx.x * 256 + threadIdx.x] = 0.f;
}

// ------------------------------------------- S[g] += Zc Zc^T (K-split+atomics)
__global__ void oni_cov(const float* __restrict__ w, const float* __restrict__ mean,
                        float* __restrict__ S) {
  __shared__ float lds[NN * APITCH];             // 128 rows x 32 k
  const int g = blockIdx.y;
  const int kbase = blockIdx.x * (DD / KSPLIT);
  const int tid = threadIdx.x;
  const int lane = tid & 31, wave = tid >> 5;
  const int M0 = wave * 16;
  const int l15 = lane & 15;
  const int kb = (lane < 16) ? 0 : 2;            // K-pair select per half-wave

  v8f acc[8];
  v8f z = {};
#pragma unroll
  for (int t = 0; t < 8; ++t) acc[t] = z;

  for (int k0 = kbase; k0 < kbase + DD / KSPLIT; k0 += KCHUNK) {
    // stage 128x32 Zc chunk: 1024 float4s, 4 per thread, b128 in/out
#pragma unroll
    for (int i = 0; i < 4; ++i) {
      int idx = i * 256 + tid;                   // float4 index
      int r = idx >> 3, kk = (idx & 7) * 4;
      int grow = g * NN + r;
      float m = mean[grow];
      float4 v = *reinterpret_cast<const float4*>(w + (size_t)grow * DD + k0 + kk);
      v.x -= m; v.y -= m; v.z -= m; v.w -= m;
      *reinterpret_cast<float4*>(&lds[r * APITCH + kk]) = v;
    }
    __syncthreads();
#pragma unroll
    for (int k4 = 0; k4 < KCHUNK; k4 += 4) {
      v2f a = *reinterpret_cast<const v2f*>(&lds[(M0 + l15) * APITCH + k4 + kb]);
      v2f bf[8];
#pragma unroll
      for (int nt = 0; nt < 8; ++nt)             // batch all B-frag loads
        bf[nt] = *reinterpret_cast<const v2f*>(&lds[(nt * 16 + l15) * APITCH + k4 + kb]);
#pragma unroll
      for (int nt = 0; nt < 8; ++nt)             // back-to-back WMMAs
        acc[nt] = wmma_f32(a, bf[nt], acc[nt]);
    }
    __syncthreads();
  }

  const int Mb = M0 + ((lane < 16) ? 0 : 8);     // C/D layout rows
  float* Sg = S + (size_t)g * NN * NN;
#pragma unroll
  for (int nt = 0; nt < 8; ++nt) {
    int col = nt * 16 + l15;
#pragma unroll
    for (int r = 0; r < 8; ++r) {
      atomicAdd(&Sg[(Mb + r) * NN + col], acc[nt][r]);
    }
  }
}

// --------------------------------- +eps*I, frobenius normalize, B0 = I, rs
__global__ void oni_norm(float* __restrict__ S, float* __restrict__ rs,
                         float* __restrict__ B0) {
  __shared__ float red[256];
  __shared__ float nrm;
  const int g = blockIdx.x;
  float* Sg = S + g * (NN * NN);
  float s = 0.f;
  for (int i = threadIdx.x; i < NN * NN; i += 256) {
    int r = i >> 7, c = i & 127;
    float v = Sg[i] + ((r == c) ? EPSV : 0.f);
    s += v * v;
  }
  red[threadIdx.x] = s;
  __syncthreads();
  for (int off = 128; off > 0; off >>= 1) {
    if (threadIdx.x < off) red[threadIdx.x] += red[threadIdx.x + off];
    __syncthreads();
  }
  if (threadIdx.x == 0) {
    float n = sqrtf(red[0]);
    nrm = n;
    rs[g] = 1.0f / sqrtf(n);                     // final scale is 1/sqrt(norm)
  }
  __syncthreads();
  float inv = 1.0f / nrm;
  float* Bg = B0 + g * (NN * NN);
  for (int i = threadIdx.x; i < NN * NN; i += 256) {
    int r = i >> 7, c = i & 127;
    Sg[i] = (Sg[i] + ((r == c) ? EPSV : 0.f)) * inv;
    Bg[i] = (r == c) ? 1.f : 0.f;
  }
}

// ------------------------- batched 128x128x128: Out = alpha*A*Bm + beta*P
__global__ void oni_gemm128(const float* __restrict__ A, const float* __restrict__ Bm,
                            const float* __restrict__ P, float* __restrict__ Out,
                            float alpha, float beta) {
  __shared__ float ldsA[NN * APITCH];
  __shared__ float ldsB[KCHUNK * NN];
  const int g = blockIdx.x;
  const float* Ag = A + g * (NN * NN);
  const float* Bg = Bm + g * (NN * NN);
  const int tid = threadIdx.x, lane = tid & 31, wave = tid >> 5;
  const int M0 = wave * 16, l15 = lane & 15;
  const int kb = (lane < 16) ? 0 : 2;

  v8f acc[8];
  v8f z = {};
#pragma unroll
  for (int t = 0; t < 8; ++t) acc[t] = z;

  for (int k0 = 0; k0 < NN; k0 += KCHUNK) {
#pragma unroll
    for (int i = 0; i < 4; ++i) {                // A chunk: [row][k], b128
      int idx = i * 256 + tid;
      int r = idx >> 3, kk = (idx & 7) * 4;
      float4 v = *reinterpret_cast<const float4*>(Ag + r * NN + k0 + kk);
      *reinterpret_cast<float4*>(&ldsA[r * APITCH + kk]) = v;
    }
#pragma unroll
    for (int i = 0; i < 4; ++i) {                // B chunk: [k][col], b128
      int idx = i * 256 + tid;
      int kk = idx >> 5, c = (idx & 31) * 4;
      float4 v = *reinterpret_cast<const float4*>(Bg + (k0 + kk) * NN + c);
      *reinterpret_cast<float4*>(&ldsB[kk * NN + c]) = v;
    }
    __syncthreads();
#pragma unroll
    for (int k4 = 0; k4 < KCHUNK; k4 += 4) {
      v2f a = *reinterpret_cast<const v2f*>(&ldsA[(M0 + l15) * APITCH + k4 + kb]);
      v2f bf[8];
#pragma unroll
      for (int nt = 0; nt < 8; ++nt) {
        int col = nt * 16 + l15;
        bf[nt].x = ldsB[(k4 + kb) * NN + col];
        bf[nt].y = ldsB[(k4 + kb + 1) * NN + col];
      }
#pragma unroll
      for (int nt = 0; nt < 8; ++nt)
        acc[nt] = wmma_f32(a, bf[nt], acc[nt]);
    }
    __syncthreads();
  }

  const int Mb = M0 + ((lane < 16) ? 0 : 8);
  float* Og = Out + g * (NN * NN);
  const float* Pg = P ? (P + g * (NN * NN)) : nullptr;
#pragma unroll
  for (int nt = 0; nt < 8; ++nt) {
    int col = nt * 16 + l15;
#pragma unroll
    for (int r = 0; r < 8; ++r) {
      float v = alpha * acc[nt][r];
      if (Pg) v += beta * Pg[(Mb + r) * NN + col];
      Og[(Mb + r) * NN + col] = v;
    }
  }
}

// ---------------------- W = (Bt @ Zc) * rs[g], Zc recomputed from weight
__global__ void oni_out(const float* __restrict__ Bt, const float* __restrict__ w,
                        const float* __restrict__ mean, const float* __restrict__ rs,
                        float* __restrict__ out) {
  __shared__ float ldsA[NN * APITCH];
  __shared__ float ldsB[KCHUNK * NN];
  const int g = blockIdx.y;
  const int N0 = blockIdx.x * 128;
  const float* Ag = Bt + g * (NN * NN);
  const int tid = threadIdx.x, lane = tid & 31, wave = tid >> 5;
  const int M0 = wave * 16, l15 = lane & 15;
  const int kb = (lane < 16) ? 0 : 2;

  v8f acc[8];
  v8f z = {};
#pragma unroll
  for (int t = 0; t < 8; ++t) acc[t] = z;

  for (int k0 = 0; k0 < NN; k0 += KCHUNK) {
#pragma unroll
    for (int i = 0; i < 4; ++i) {                // A = Bt chunk, b128
      int idx = i * 256 + tid;
      int r = idx >> 3, kk = (idx & 7) * 4;
      float4 v = *reinterpret_cast<const float4*>(Ag + r * NN + k0 + kk);
      *reinterpret_cast<float4*>(&ldsA[r * APITCH + kk]) = v;
    }
#pragma unroll
    for (int i = 0; i < 4; ++i) {                // B[k][col] = Zc[k][N0+col]
      int idx = i * 256 + tid;
      int kk = idx >> 5, c = (idx & 31) * 4;
      int grow = g * NN + k0 + kk;
      float m = mean[grow];
      float4 v = *reinterpret_cast<const float4*>(w + (size_t)grow * DD + N0 + c);
      v.x -= m; v.y -= m; v.z -= m; v.w -= m;
      *reinterpret_cast<float4*>(&ldsB[kk * NN + c]) = v;
    }
    __syncthreads();
#pragma unroll
    for (int k4 = 0; k4 < KCHUNK; k4 += 4) {
      v2f a = *reinterpret_cast<const v2f*>(&ldsA[(M0 + l15) * APITCH + k4 + kb]);
      v2f bf[8];
#pragma unroll
      for (int nt = 0; nt < 8; ++nt) {
        int col = nt * 16 + l15;
        bf[nt].x = ldsB[(k4 + kb) * NN + col];
        bf[nt].y = ldsB[(k4 + kb + 1) * NN + col];
      }
#pragma unroll
      for (int nt = 0; nt < 8; ++nt)
        acc[nt] = wmma_f32(a, bf[nt], acc[nt]);
    }
    __syncthreads();
  }

  const float scale = rs[g];
  const int Mb = M0 + ((lane < 16) ? 0 : 8);
#pragma unroll
  for (int nt = 0; nt < 8; ++nt) {
    int col = N0 + nt * 16 + l15;
#pragma unroll
    for (int r = 0; r < 8; ++r) {
      out[(size_t)(g * NN + Mb + r) * DD + col] = acc[nt][r] * scale;
    }
  }
}

extern "C" void kernel_launch(void* const* d_in, const int* in_sizes, int n_in,
                              void* d_out, int out_size, void* d_ws, size_t ws_size,
                              hipStream_t stream) {
  const float* w = (const float*)d_in[0];
  float* out = (float*)d_out;
  float* ws = (float*)d_ws;

  // workspace layout (floats): means | rs | pad | S | BA | BB | T2 | T3
  float* mean = ws;                 // 2048
  float* rs = ws + 2048;            // 16
  float* S  = ws + 2080;            // 16*128*128
  float* BA = S + MATF;
  float* BB = BA + MATF;
  float* T2 = BB + MATF;
  float* T3 = T2 + MATF;

  oni_mean<<<NG * NN, 256, 0, stream>>>(w, mean);
  oni_zero<<<MATF / 256, 256, 0, stream>>>(S);
  oni_cov<<<dim3(KSPLIT, NG), 256, 0, stream>>>(w, mean, S);
  oni_norm<<<NG, 256, 0, stream>>>(S, rs, BA);

  float* Bc = BA;
  float* Bn = BB;
  for (int t = 0; t < TITER; ++t) {
    oni_gemm128<<<NG, 256, 0, stream>>>(Bc, Bc, nullptr, T2, 1.f, 0.f);  // B^2
    oni_gemm128<<<NG, 256, 0, stream>>>(T2, Bc, nullptr, T3, 1.f, 0.f);  // B^3
    // Bn = 1.5*B - 0.5*(B^3 @ S)
    oni_gemm128<<<NG, 256, 0, stream>>>(T3, S, Bc, Bn, -0.5f, 1.5f);
    float* tmp = Bc; Bc = Bn; Bn = tmp;
  }

  oni_out<<<dim3(DD / 128, NG), 256, 0, stream>>>(Bc, w, mean, rs, out);
}